// TokenUOTBlock_35983236006597
// MI455X (gfx1250) — compile-verified
//
#include <hip/hip_runtime.h>
#include <hip/hip_bf16.h>
#include <math.h>

#define NTOK 1024
#define DMODEL 512
#define NHEAD 16
#define DHEAD 32
#define DFF 2048
#define KSTAGE 16                     // K rows staged into LDS per double-buffer slot

typedef __attribute__((ext_vector_type(2))) float v2f;
typedef __attribute__((ext_vector_type(8))) float v8f;

// ---------------------------------------------------------------------------
// LayerNorm over D=512 with affine, one block (256 thr) per row
// ---------------------------------------------------------------------------
__global__ void ln_kernel(const float* __restrict__ x, const float* __restrict__ w,
                          const float* __restrict__ b, float* __restrict__ y) {
    __shared__ float red0[8], red1[8];
    const int n = blockIdx.x;
    const int tid = threadIdx.x;                       // 0..255
    const float* xr = x + (size_t)n * DMODEL;
    float v0 = xr[tid], v1 = xr[tid + 256];
    float s = v0 + v1, ss = v0 * v0 + v1 * v1;
    for (int off = 16; off > 0; off >>= 1) {
        s  += __shfl_xor(s, off);
        ss += __shfl_xor(ss, off);
    }
    const int wave = tid >> 5, lane = tid & 31;
    if (lane == 0) { red0[wave] = s; red1[wave] = ss; }
    __syncthreads();
    if (wave == 0) {
        s  = (lane < 8) ? red0[lane] : 0.0f;
        ss = (lane < 8) ? red1[lane] : 0.0f;
        for (int off = 4; off > 0; off >>= 1) {
            s += __shfl_xor(s, off); ss += __shfl_xor(ss, off);
        }
        if (lane == 0) { red0[0] = s; red1[0] = ss; }
    }
    __syncthreads();
    const float mean = red0[0] / (float)DMODEL;
    const float var  = red1[0] / (float)DMODEL - mean * mean;
    const float r = rsqrtf(var + 1e-5f);
    float* yr = y + (size_t)n * DMODEL;
    yr[tid]       = (v0 - mean) * r * w[tid]       + b[tid];
    yr[tid + 256] = (v1 - mean) * r * w[tid + 256] + b[tid + 256];
}

// ---------------------------------------------------------------------------
// Per-head LN over DH=32 (no affine): one wave32 per (n,h) slice
// ---------------------------------------------------------------------------
__global__ void head_ln_kernel(const float* __restrict__ X, float* __restrict__ Y) {
    const int wave = threadIdx.x >> 5, lane = threadIdx.x & 31;
    const int idx = blockIdx.x * 8 + wave;             // over N*H
    const int n = idx >> 4, h = idx & 15;
    const size_t off = (size_t)n * DMODEL + h * DHEAD + lane;
    float v = X[off];
    float s = v, ss = v * v;
    for (int o = 16; o > 0; o >>= 1) {
        s += __shfl_xor(s, o); ss += __shfl_xor(ss, o);
    }
    const float mean = s / 32.0f;
    const float var  = ss / 32.0f - mean * mean;
    Y[off] = (v - mean) * rsqrtf(var + 1e-5f);
}

// ---------------------------------------------------------------------------
// Generic f32 WMMA GEMM: C[M x Nd] = A[M x K] @ B[K x Nd] (+ optional R)
// One wave32 per block; wave computes a 16x64 output tile (4 accumulators).
// B K-panels (KSTAGE x 64) are bulk-staged into LDS with
// GLOBAL_LOAD_ASYNC_TO_LDS_B128 (ASYNCcnt, double-buffered, s_wait_asynccnt),
// WMMA B fragments then come from ds_load.  A stays as direct b64 loads.
// grid = (M/16, Nd/64)
// ---------------------------------------------------------------------------
__global__ void gemm_f32_wmma(const float* __restrict__ A, const float* __restrict__ B,
                              const float* __restrict__ R, float* __restrict__ C,
                              int Md, int Kd, int Nd) {
    __shared__ float bsm[2][KSTAGE * 64];              // 2 x 4KB
    const int lane = threadIdx.x;
    const int half = lane >> 4;          // 0: K pair {0,1}, 1: K pair {2,3}
    const int l16  = lane & 15;
    const int row0 = blockIdx.x * 16;
    const int col0 = blockIdx.y * 64;

    // per-lane coords for async staging: 32 lanes x 16B = 2 rows of 64 floats
    const int srow = lane >> 4;                        // 0/1: row within pair
    const int scol = (lane & 15) * 4;                  // float col (16B chunk)
    const uint32_t lds_base = (uint32_t)(uintptr_t)(&bsm[0][0]);

    // stage KSTAGE x 64 panel of B (rows k0..k0+KSTAGE-1) into buffer `buf`
    auto stage = [&](int k0, int buf) {
        const uint32_t lb = lds_base + (uint32_t)buf * (KSTAGE * 64 * 4);
        #pragma unroll
        for (int i = 0; i < KSTAGE / 2; ++i) {
            const int r = 2 * i + srow;
            const uint64_t gaddr =
                (uint64_t)(uintptr_t)(B + (size_t)(k0 + r) * Nd + col0 + scol);
            const uint32_t laddr = lb + (uint32_t)(r * 64 + scol) * 4;
            asm volatile("global_load_async_to_lds_b128 %0, %1, off"
                         :: "v"(laddr), "v"(gaddr) : "memory");
        }
    };

    v8f acc0 = {}, acc1 = {}, acc2 = {}, acc3 = {};
    const float* arow = A + (size_t)(row0 + l16) * Kd + half * 2;

    stage(0, 0);
    int buf = 0;
    for (int k0 = 0; k0 < Kd; k0 += KSTAGE) {
        if (k0 + KSTAGE < Kd) {
            stage(k0 + KSTAGE, buf ^ 1);
            // older 8 transfers (current buffer) complete; 8 newest in flight
            asm volatile("s_wait_asynccnt 0x8" ::: "memory");
        } else {
            asm volatile("s_wait_asynccnt 0x0" ::: "memory");
        }
        const float* bs = &bsm[buf][0];
        #pragma unroll
        for (int kk = 0; kk < KSTAGE; kk += 4) {
            v2f a;
            a.x = arow[k0 + kk];
            a.y = arow[k0 + kk + 1];
            const float* bp = bs + (kk + half * 2) * 64 + l16;   // LDS reads
            v2f b0, b1, b2, b3;
            b0.x = bp[0];  b0.y = bp[64];
            b1.x = bp[16]; b1.y = bp[64 + 16];
            b2.x = bp[32]; b2.y = bp[64 + 32];
            b3.x = bp[48]; b3.y = bp[64 + 48];
            acc0 = __builtin_amdgcn_wmma_f32_16x16x4_f32(false, a, false, b0, (short)0, acc0, false, false);
            acc1 = __builtin_amdgcn_wmma_f32_16x16x4_f32(false, a, false, b1, (short)0, acc1, false, false);
            acc2 = __builtin_amdgcn_wmma_f32_16x16x4_f32(false, a, false, b2, (short)0, acc2, false, false);
            acc3 = __builtin_amdgcn_wmma_f32_16x16x4_f32(false, a, false, b3, (short)0, acc3, false, false);
        }
        buf ^= 1;
    }

    #pragma unroll
    for (int r = 0; r < 8; ++r) {
        const int row = row0 + r + half * 8;
        float* cp = C + (size_t)row * Nd + col0 + l16;
        if (R) {
            const float* rp = R + (size_t)row * Nd + col0 + l16;
            cp[0]  = acc0[r] + rp[0];
            cp[16] = acc1[r] + rp[16];
            cp[32] = acc2[r] + rp[32];
            cp[48] = acc3[r] + rp[48];
        } else {
            cp[0]  = acc0[r];
            cp[16] = acc1[r];
            cp[32] = acc2[r];
            cp[48] = acc3[r];
        }
    }
}

// ---------------------------------------------------------------------------
// Fused content + geometry + M build:
//   content = -(Q_ln . K_ln)/sqrt(DH)
//   C = content + w_rel + geo_gate[h]*w_dist[h]*f_dist(i,j)
//   M = -C / eps_h
// One wave32 per 16x16 (i,j) tile per head; WMMA over DH=32 (8 K-steps).
// grid = (N/16, N/16, H)
// ---------------------------------------------------------------------------
__global__ void content_M_kernel(const float* __restrict__ Qln, const float* __restrict__ Kln,
                                 const float* __restrict__ wrel, const float* __restrict__ xres,
                                 const float* __restrict__ wdist, const float* __restrict__ geo,
                                 float* __restrict__ M) {
    const int lane = threadIdx.x;
    const int half = lane >> 4, l16 = lane & 15;
    const int i0 = blockIdx.x * 16, j0 = blockIdx.y * 16, h = blockIdx.z;
    const float eps = 0.5f * (float)(1 << (h >> 2));   // [0.5,1,2,4] by quad
    const float neg_inv_eps = -1.0f / eps;
    const float effw = geo[h] * wdist[h];
    v8f acc = {};
    const float* aq = Qln + (size_t)(i0 + l16) * DMODEL + h * DHEAD + half * 2;
    const float* bk = Kln + (size_t)(j0 + l16) * DMODEL + h * DHEAD + half * 2;
    #pragma unroll
    for (int k0 = 0; k0 < DHEAD; k0 += 4) {
        v2f a, b;
        a.x = aq[k0]; a.y = aq[k0 + 1];
        b.x = bk[k0]; b.y = bk[k0 + 1];          // B = Kln^T: element (k, j)
        acc = __builtin_amdgcn_wmma_f32_16x16x4_f32(false, a, false, b, (short)0, acc, false, false);
    }
    const float nscale = -0.1767766952966369f;   // -1/sqrt(32)
    const int j = j0 + l16;
    const float xj0 = xres[j * 3 + 0], xj1 = xres[j * 3 + 1], xj2 = xres[j * 3 + 2];
    #pragma unroll
    for (int r = 0; r < 8; ++r) {
        const int i = i0 + r + half * 8;
        const float dx = xres[i * 3 + 0] - xj0;
        const float dy = xres[i * 3 + 1] - xj1;
        const float dz = xres[i * 3 + 2] - xj2;
        const float dist = sqrtf(fmaxf(dx * dx + dy * dy + dz * dz, 1e-12f));
        const float fd = dist / (10.0f + dist);
        const size_t off = ((size_t)h * NTOK + i) * NTOK + j;
        const float C = nscale * acc[r] + wrel[off] + effw * fd;
        M[off] = C * neg_inv_eps;
    }
}

// ---------------------------------------------------------------------------
// log(clip(mu/nu, 1e-8)) precompute
// ---------------------------------------------------------------------------
__global__ void log_clip_kernel(const float* __restrict__ mu, const float* __restrict__ nu,
                                float* __restrict__ lmu, float* __restrict__ lnu, int n) {
    const int i = blockIdx.x * blockDim.x + threadIdx.x;
    if (i < n) {
        lmu[i] = __logf(fmaxf(mu[i], 1e-8f));
        lnu[i] = __logf(fmaxf(nu[i], 1e-8f));
    }
}

// ---------------------------------------------------------------------------
// Sinkhorn row half: lu[h,i] = fac * (logmu - logsumexp_j(M[h,i,j] + lv[h,j]))
// One wave32 per row; streaming (max,sum) logsumexp, shuffle-combined.
// grid = H*N/8, block 256
// ---------------------------------------------------------------------------
__global__ void sinkhorn_row_kernel(const float* __restrict__ M, const float* __restrict__ lv,
                                    const float* __restrict__ logmu, float* __restrict__ lu) {
    const int wave = threadIdx.x >> 5, lane = threadIdx.x & 31;
    const int row = blockIdx.x * 8 + wave;             // over H*N
    const int h = row >> 10;
    const float* Mr = M + (size_t)row * NTOK;
    const float* lvh = lv + (h << 10);
    float m = -INFINITY, s = 0.0f;
    for (int j = lane; j < NTOK; j += 32) {
        __builtin_prefetch(&Mr[j + 256], 0, 0);        // global_prefetch_b8, L2-resident stream
        const float v = Mr[j] + lvh[j];
        if (v > m) { s = s * __expf(m - v) + 1.0f; m = v; }
        else       { s += __expf(v - m); }
    }
    for (int off = 16; off > 0; off >>= 1) {
        const float m2 = __shfl_xor(m, off);
        const float s2 = __shfl_xor(s, off);
        if (m2 > m) { s = s * __expf(m - m2) + s2; m = m2; }
        else        { s += s2 * __expf(m2 - m); }
    }
    if (lane == 0) {
        const float eps = 0.5f * (float)(1 << (h >> 2));
        const float fac = 1.0f / (1.0f + eps);
        lu[row] = fac * (logmu[row] - (m + __logf(s)));
    }
}

// ---------------------------------------------------------------------------
// Sinkhorn column half: lv[h,j] = fac*(lognu - logsumexp_i(M[h,i,j] + lu[h,i]))
// Block: 256 thr = 64 columns x 4 row-groups (coalesced across j), LDS combine.
// grid = (N/64, H)
// ---------------------------------------------------------------------------
__global__ void sinkhorn_col_kernel(const float* __restrict__ M, const float* __restrict__ lu,
                                    const float* __restrict__ lognu, float* __restrict__ lv) {
    __shared__ float sm[4][64], ss[4][64];
    const int tid = threadIdx.x;
    const int c = tid & 63, g = tid >> 6;
    const int j0 = blockIdx.x * 64, h = blockIdx.y;
    const int j = j0 + c;
    const float* Mh = M + (size_t)h * NTOK * NTOK;
    const float* luh = lu + (h << 10);
    float m = -INFINITY, s = 0.0f;
    const int iend = (g + 1) * 256;
    for (int i = g * 256; i < iend; ++i) {
        __builtin_prefetch(&Mh[(size_t)(i + 4) * NTOK + j], 0, 0);
        const float v = Mh[(size_t)i * NTOK + j] + luh[i];
        if (v > m) { s = s * __expf(m - v) + 1.0f; m = v; }
        else       { s += __expf(v - m); }
    }
    sm[g][c] = m; ss[g][c] = s;
    __syncthreads();
    if (tid < 64) {
        float M0 = sm[0][c], S0 = ss[0][c];
        #pragma unroll
        for (int gg = 1; gg < 4; ++gg) {
            const float m2 = sm[gg][c], s2 = ss[gg][c];
            if (m2 > M0) { S0 = S0 * __expf(M0 - m2) + s2; M0 = m2; }
            else         { S0 += s2 * __expf(m2 - M0); }
        }
        const float eps = 0.5f * (float)(1 << (h >> 2));
        const float fac = 1.0f / (1.0f + eps);
        lv[(h << 10) + j] = fac * (lognu[(h << 10) + j] - (M0 + __logf(S0)));
    }
}

// ---------------------------------------------------------------------------
// Fused: rowsum[h,i] = sum_j exp(M+lu+lv); xcent[h,i,:] = (sum_j T*x_res)/rowsum
// One wave32 per row.  grid = H*N/8, block 256
// ---------------------------------------------------------------------------
__global__ void rowsum_xcent_kernel(const float* __restrict__ M, const float* __restrict__ lu,
                                    const float* __restrict__ lv, const float* __restrict__ xres,
                                    float* __restrict__ rowsum, float* __restrict__ xcent) {
    const int wave = threadIdx.x >> 5, lane = threadIdx.x & 31;
    const int row = blockIdx.x * 8 + wave;             // over H*N
    const int h = row >> 10;
    const float* Mr = M + (size_t)row * NTOK;
    const float* lvh = lv + (h << 10);
    const float lui = lu[row];
    float S = 0.0f, Sx = 0.0f, Sy = 0.0f, Sz = 0.0f;
    for (int j = lane; j < NTOK; j += 32) {
        const float t = __expf(Mr[j] + lui + lvh[j]);
        S += t;
        Sx += t * xres[j * 3 + 0];
        Sy += t * xres[j * 3 + 1];
        Sz += t * xres[j * 3 + 2];
    }
    for (int off = 16; off > 0; off >>= 1) {
        S  += __shfl_xor(S, off);
        Sx += __shfl_xor(Sx, off);
        Sy += __shfl_xor(Sy, off);
        Sz += __shfl_xor(Sz, off);
    }
    if (lane == 0) {
        rowsum[row] = S;
        const float inv = 1.0f / (S + 1e-8f);
        xcent[(size_t)row * 3 + 0] = Sx * inv;
        xcent[(size_t)row * 3 + 1] = Sy * inv;
        xcent[(size_t)row * 3 + 2] = Sz * inv;
    }
}

// ---------------------------------------------------------------------------
// T_norm @ V via WMMA: T recomputed on the fly from M+lu+lv (no second 67MB
// tensor).  Wave computes 16 rows x 32 cols (full DH) for one head.
// grid = (N/16, H)
// ---------------------------------------------------------------------------
__global__ void tv_wmma_kernel(const float* __restrict__ M, const float* __restrict__ lu,
                               const float* __restrict__ lv, const float* __restrict__ rowsum,
                               const float* __restrict__ Vmat, float* __restrict__ TV) {
    const int lane = threadIdx.x;
    const int half = lane >> 4, l16 = lane & 15;
    const int i0 = blockIdx.x * 16, h = blockIdx.y;
    const float* Mh = M + (size_t)h * NTOK * NTOK;
    const float* lvh = lv + (h << 10);
    const int ia = i0 + l16;                           // fixed A-row per lane
    const float lui = lu[(h << 10) + ia];
    const float inv_rs = 1.0f / (rowsum[(h << 10) + ia] + 1e-8f);
    const float* Mrow = Mh + (size_t)ia * NTOK;
    v8f acc0 = {}, acc1 = {};
    for (int k0 = 0; k0 < NTOK; k0 += 4) {
        const int ka = k0 + half * 2;
        v2f a;
        a.x = __expf(Mrow[ka]     + lui + lvh[ka])     * inv_rs;
        a.y = __expf(Mrow[ka + 1] + lui + lvh[ka + 1]) * inv_rs;
        const float* bp = Vmat + (size_t)ka * DMODEL + h * DHEAD + l16;
        v2f b0, b1;
        b0.x = bp[0];  b0.y = bp[DMODEL];
        b1.x = bp[16]; b1.y = bp[DMODEL + 16];
        acc0 = __builtin_amdgcn_wmma_f32_16x16x4_f32(false, a, false, b0, (short)0, acc0, false, false);
        acc1 = __builtin_amdgcn_wmma_f32_16x16x4_f32(false, a, false, b1, (short)0, acc1, false, false);
    }
    #pragma unroll
    for (int r = 0; r < 8; ++r) {
        const int i = i0 + r + half * 8;
        float* tp = TV + ((size_t)(h << 10) + i) * DHEAD + l16;
        tp[0]  = acc0[r];
        tp[16] = acc1[r];
    }
}

// ---------------------------------------------------------------------------
// o[n, h*32+d] = sigmoid(G[n, h*32+d]) * TV[h, n, d]
// ---------------------------------------------------------------------------
__global__ void o_assemble_kernel(const float* __restrict__ G, const float* __restrict__ TV,
                                  float* __restrict__ o) {
    const int idx = blockIdx.x * blockDim.x + threadIdx.x;     // N*D
    const int n = idx >> 9, col = idx & 511;
    const int h = col >> 5, d = col & 31;
    const float g = G[idx];
    const float sig = 1.0f / (1.0f + __expf(-g));
    o[idx] = sig * TV[((size_t)(h << 10) + n) * DHEAD + d];
}

// ---------------------------------------------------------------------------
// x_res update: x += sum_h gamma[h] * (x - xcent[h,n,:])
// ---------------------------------------------------------------------------
__global__ void x_update_kernel(const float* __restrict__ x, const float* __restrict__ xcent,
                                const float* __restrict__ gamma, float* __restrict__ xout) {
    const int idx = blockIdx.x * blockDim.x + threadIdx.x;     // N*3
    if (idx >= NTOK * 3) return;
    const int n = idx / 3, c = idx % 3;
    const float xv = x[idx];
    float acc = 0.0f;
    #pragma unroll
    for (int h = 0; h < NHEAD; ++h)
        acc += gamma[h] * (xv - xcent[((size_t)(h << 10) + n) * 3 + c]);
    xout[idx] = xv + acc;
}

// ---------------------------------------------------------------------------
// SwiGLU activation, in place on u1: u1 = silu(u1) * u3
// ---------------------------------------------------------------------------
__global__ void act_kernel(float* __restrict__ u1, const float* __restrict__ u3) {
    const int idx = blockIdx.x * blockDim.x + threadIdx.x;     // N*DFF
    const float u = u1[idx];
    u1[idx] = (u / (1.0f + __expf(-u))) * u3[idx];
}

// ===========================================================================
extern "C" void kernel_launch(void* const* d_in, const int* in_sizes, int n_in,
                              void* d_out, int out_size, void* d_ws, size_t ws_size,
                              hipStream_t stream) {
    const float* h_in  = (const float*)d_in[0];
    const float* x_res = (const float*)d_in[1];
    const float* mu    = (const float*)d_in[2];
    const float* nu    = (const float*)d_in[3];
    const float* lu0   = (const float*)d_in[4];
    const float* lv0   = (const float*)d_in[5];
    const float* wrel  = (const float*)d_in[6];
    const float* wdist = (const float*)d_in[7];
    /* d_in[8] = pos_bins (unused by reference) */
    const float* geo   = (const float*)d_in[9];
    const float* lnw   = (const float*)d_in[10];
    const float* lnb   = (const float*)d_in[11];
    const float* Wq    = (const float*)d_in[12];
    const float* Wk    = (const float*)d_in[13];
    const float* Wv    = (const float*)d_in[14];
    const float* Wg    = (const float*)d_in[15];
    const float* Wo    = (const float*)d_in[16];
    const float* gamma = (const float*)d_in[17];
    const float* lnfw  = (const float*)d_in[18];
    const float* lnfb  = (const float*)d_in[19];
    const float* W1    = (const float*)d_in[20];
    const float* W3    = (const float*)d_in[21];
    const float* W2    = (const float*)d_in[22];

    float* out = (float*)d_out;
    float* out_h  = out;                     // N*D
    float* out_x  = out + 524288;            // N*3
    float* out_lu = out + 527360;            // H*N
    float* out_lv = out + 543744;            // H*N

    // workspace layout (floats)
    float* ws   = (float*)d_ws;
    float* hn   = ws;                        // 524288
    float* Qm   = hn   + 524288;
    float* Km   = Qm   + 524288;
    float* Vm   = Km   + 524288;
    float* Gm   = Vm   + 524288;
    float* Qln  = Gm   + 524288;
    float* Kln  = Qln  + 524288;
    float* lu   = Kln  + 524288;             // 16384
    float* lv   = lu   + 16384;
    float* lmu  = lv   + 16384;
    float* lnu  = lmu  + 16384;
    float* rsum = lnu  + 16384;
    float* xc   = rsum + 16384;              // H*N*3 = 49152
    float* TV   = xc   + 49152;              // 524288
    float* ob   = TV   + 524288;
    float* h2   = ob   + 524288;
    float* ffb  = h2   + 524288;
    float* u1   = ffb  + 524288;             // N*DFF = 2097152
    float* u3   = u1   + 2097152;
    float* Mb   = u3   + 2097152;            // H*N*N = 16777216

    const dim3 wave(32);

    // 1) attention LN + projections
    ln_kernel<<<NTOK, 256, 0, stream>>>(h_in, lnw, lnb, hn);
    const dim3 g512(NTOK / 16, DMODEL / 64);
    gemm_f32_wmma<<<g512, wave, 0, stream>>>(hn, Wq, nullptr, Qm, NTOK, DMODEL, DMODEL);
    gemm_f32_wmma<<<g512, wave, 0, stream>>>(hn, Wk, nullptr, Km, NTOK, DMODEL, DMODEL);
    gemm_f32_wmma<<<g512, wave, 0, stream>>>(hn, Wv, nullptr, Vm, NTOK, DMODEL, DMODEL);
    gemm_f32_wmma<<<g512, wave, 0, stream>>>(hn, Wg, nullptr, Gm, NTOK, DMODEL, DMODEL);
    head_ln_kernel<<<NTOK * NHEAD / 8, 256, 0, stream>>>(Qm, Qln);
    head_ln_kernel<<<NTOK * NHEAD / 8, 256, 0, stream>>>(Km, Kln);

    // 2) cost tensor M (only full-size HBM write; stays L2-resident after)
    content_M_kernel<<<dim3(NTOK / 16, NTOK / 16, NHEAD), wave, 0, stream>>>(
        Qln, Kln, wrel, x_res, wdist, geo, Mb);

    // 3) Sinkhorn
    log_clip_kernel<<<(NHEAD * NTOK + 255) / 256, 256, 0, stream>>>(mu, nu, lmu, lnu, NHEAD * NTOK);
    hipMemcpyAsync(lu, lu0, NHEAD * NTOK * sizeof(float), hipMemcpyDeviceToDevice, stream);
    hipMemcpyAsync(lv, lv0, NHEAD * NTOK * sizeof(float), hipMemcpyDeviceToDevice, stream);
    for (int it = 0; it < 20; ++it) {
        sinkhorn_row_kernel<<<NHEAD * NTOK / 8, 256, 0, stream>>>(Mb, lv, lmu, lu);
        sinkhorn_col_kernel<<<dim3(NTOK / 64, NHEAD), 256, 0, stream>>>(Mb, lu, lnu, lv);
    }
    hipMemcpyAsync(out_lu, lu, NHEAD * NTOK * sizeof(float), hipMemcpyDeviceToDevice, stream);
    hipMemcpyAsync(out_lv, lv, NHEAD * NTOK * sizeof(float), hipMemcpyDeviceToDevice, stream);

    // 4) transport application
    rowsum_xcent_kernel<<<NHEAD * NTOK / 8, 256, 0, stream>>>(Mb, lu, lv, x_res, rsum, xc);
    tv_wmma_kernel<<<dim3(NTOK / 16, NHEAD), wave, 0, stream>>>(Mb, lu, lv, rsum, Vm, TV);
    o_assemble_kernel<<<NTOK * DMODEL / 256, 256, 0, stream>>>(Gm, TV, ob);
    gemm_f32_wmma<<<g512, wave, 0, stream>>>(ob, Wo, h_in, h2, NTOK, DMODEL, DMODEL);
    x_update_kernel<<<(NTOK * 3 + 255) / 256, 256, 0, stream>>>(x_res, xc, gamma, out_x);

    // 5) FFN
    ln_kernel<<<NTOK, 256, 0, stream>>>(h2, lnfw, lnfb, ffb);
    const dim3 gff(NTOK / 16, DFF / 64);
    gemm_f32_wmma<<<gff, wave, 0, stream>>>(ffb, W1, nullptr, u1, NTOK, DMODEL, DFF);
    gemm_f32_wmma<<<gff, wave, 0, stream>>>(ffb, W3, nullptr, u3, NTOK, DMODEL, DFF);
    act_kernel<<<NTOK * DFF / 256, 256, 0, stream>>>(u1, u3);
    gemm_f32_wmma<<<g512, wave, 0, stream>>>(u1, W2, h2, out_h, NTOK, DFF, DMODEL);
}